// SMPLRenderer_10350871183721
// MI455X (gfx1250) — compile-verified
//
#include <hip/hip_runtime.h>
#include <hip/hip_bf16.h>

// ---------------- problem constants ----------------
constexpr int BB   = 4;
constexpr int NV   = 6890;
constexpr int NF   = 27552;          // 2 * 13776
constexpr int FPAD = 27648;          // padded to multiple of 128 (and 16)
constexpr int IMG  = 128;
constexpr int UVH  = 256, UVW = 256;
constexpr int NCHUNK = FPAD / 16;            // 1728 chunks of 16 faces
constexpr int CH_PER_STAGE = 16;             // 16 chunks staged per LDS buffer
constexpr int NSTAGE = NCHUNK / CH_PER_STAGE;// 108
constexpr float FLEN = 500.0f, PNEAR = 0.1f, PFAR = 25.0f, FBIG = 1.0e10f;

// ---------------- workspace layout (floats) ----------------
constexpr size_t OFF_TEX  = 0;                                        // B*NF*9*3
constexpr size_t OFF_FV   = OFF_TEX  + (size_t)BB * NF * 9 * 3;       // B*NF*9
constexpr size_t OFF_COEF = OFF_FV   + (size_t)BB * NF * 9;           // B*NCHUNK*256
constexpr size_t OFF_AREA = OFF_COEF + (size_t)BB * NCHUNK * 256;     // B*NCHUNK*16
constexpr size_t OFF_FIM  = OFF_AREA + (size_t)BB * NCHUNK * 16;      // B*IMG*IMG ints

// ---------------- CDNA5 vector types ----------------
typedef float v2f __attribute__((ext_vector_type(2)));
typedef float v8f __attribute__((ext_vector_type(8)));
typedef int   v4i __attribute__((vector_size(16)));
typedef __attribute__((address_space(1))) v4i gv4i;   // global int4
typedef __attribute__((address_space(3))) v4i lv4i;   // LDS int4

#if __has_builtin(__builtin_amdgcn_global_load_async_to_lds_b128)
#define HAVE_ASYNC 1
#else
#define HAVE_ASYNC 0
#endif

#if HAVE_ASYNC
#if __has_builtin(__builtin_amdgcn_s_wait_asynccnt)
#define WAIT_ASYNC() __builtin_amdgcn_s_wait_asynccnt(0)
#else
#define WAIT_ASYNC() asm volatile("s_wait_asynccnt 0" ::: "memory")
#endif
#else
#define WAIT_ASYNC() do {} while (0)
#endif

__device__ __forceinline__ void cp16(const float* g, float* l) {
#if HAVE_ASYNC
  __builtin_amdgcn_global_load_async_to_lds_b128((gv4i*)(g), (lv4i*)(l), 0, 0);
#else
  *(float4*)l = *(const float4*)g;
#endif
}

__device__ __forceinline__ v8f wmma4(v2f a, v2f b) {
  v8f c = {};
  return __builtin_amdgcn_wmma_f32_16x16x4_f32(false, a, false, b, (short)0, c,
                                               false, false);
}

// ============ kernel 1: bilinear texture extraction (lighting == identity) ====
__global__ void k_tex(const float* __restrict__ uv, const float* __restrict__ samp,
                      float* __restrict__ tex) {
  int tid = blockIdx.x * blockDim.x + threadIdx.x;
  if (tid >= BB * NF * 9) return;
  int b = tid / (NF * 9);
  int rem = tid - b * (NF * 9);            // f*9 + k
  float gx = samp[rem * 2 + 0], gy = samp[rem * 2 + 1];
  float x = (gx + 1.0f) * (UVW * 0.5f) - 0.5f;
  float y = (gy + 1.0f) * (UVH * 0.5f) - 0.5f;
  float x0f = floorf(x), y0f = floorf(y);
  float wx = x - x0f, wy = y - y0f;
  int x0 = (int)x0f, y0 = (int)y0f;
  const float* img = uv + (size_t)b * 3 * UVH * UVW;
  float a0 = 0.f, a1 = 0.f, a2 = 0.f;
  auto gat = [&](int yi, int xi, float w) {
    bool valid = (xi >= 0) && (xi < UVW) && (yi >= 0) && (yi < UVH);
    int cy = min(max(yi, 0), UVH - 1), cx = min(max(xi, 0), UVW - 1);
    int idx = cy * UVW + cx;
    float wv = valid ? w : 0.0f;
    a0 += img[0 * UVH * UVW + idx] * wv;
    a1 += img[1 * UVH * UVW + idx] * wv;
    a2 += img[2 * UVH * UVW + idx] * wv;
  };
  gat(y0,     x0,     (1.f - wx) * (1.f - wy));
  gat(y0,     x0 + 1, wx * (1.f - wy));
  gat(y0 + 1, x0,     (1.f - wx) * wy);
  gat(y0 + 1, x0 + 1, wx * wy);
  float* o = tex + ((size_t)b * NF * 9 + rem) * 3;
  o[0] = a0; o[1] = a1; o[2] = a2;
}

// ============ kernel 2: projection + edge-function coefficients ==============
// Per face, quantities q0..q3 = (w0, w1, w2, sum_i w_i/z_i), each affine:
//   value(px,py) = A + Bx*px + Cy*py.  A-matrix lane layout for 16x16x4 f32 WMMA:
//   lane m (m<16):   (K0,K1) = (A, Bx);  lane 16+m: (K2,K3) = (Cy, 0).
__global__ void k_face(const float* __restrict__ cam, const float* __restrict__ verts,
                       const int* __restrict__ faces, float* __restrict__ coef,
                       float* __restrict__ areab, float* __restrict__ fvbuf) {
  int tid = blockIdx.x * blockDim.x + threadIdx.x;
  if (tid >= BB * FPAD) return;
  int b = tid / FPAD;
  int f = tid - b * FPAD;
  int chunk = f >> 4, m = f & 15;

  float A0=0,B0=0,C0=0,A1=0,B1=0,C1=0,A2=0,B2=0,C2=0,Aq=0,Bq=0,Cq=0,area=0;
  if (f < NF) {
    int i0 = faces[f * 3 + 0], i1 = faces[f * 3 + 1], i2 = faces[f * 3 + 2];
    float s  = cam[b * 3 + 0];
    float ccx = cam[b * 3 + 1], ccy = cam[b * 3 + 2];
    float tz = FLEN / ((IMG * 0.5f) * s);
    const float* vb = verts + (size_t)b * NV * 3;
    float x0 = s * (vb[i0 * 3 + 0] + ccx), y0 = s * (vb[i0 * 3 + 1] + ccy), z0 = vb[i0 * 3 + 2] + tz;
    float x1 = s * (vb[i1 * 3 + 0] + ccx), y1 = s * (vb[i1 * 3 + 1] + ccy), z1 = vb[i1 * 3 + 2] + tz;
    float x2 = s * (vb[i2 * 3 + 0] + ccx), y2 = s * (vb[i2 * 3 + 1] + ccy), z2 = vb[i2 * 3 + 2] + tz;
    float* fv = fvbuf + (size_t)(b * NF + f) * 9;
    fv[0]=x0; fv[1]=y0; fv[2]=z0; fv[3]=x1; fv[4]=y1; fv[5]=z1; fv[6]=x2; fv[7]=y2; fv[8]=z2;

    B0 = -(y2 - y1); C0 = (x2 - x1); A0 = (y2 - y1) * x1 - (x2 - x1) * y1;
    B1 = -(y0 - y2); C1 = (x0 - x2); A1 = (y0 - y2) * x2 - (x0 - x2) * y2;
    B2 = -(y1 - y0); C2 = (x1 - x0); A2 = (y1 - y0) * x0 - (x1 - x0) * y0;
    area = A0 + A1 + A2;                       // linear terms cancel analytically
    float iz0 = 1.0f / z0, iz1 = 1.0f / z1, iz2 = 1.0f / z2;
    Aq = A0 * iz0 + A1 * iz1 + A2 * iz2;
    Bq = B0 * iz0 + B1 * iz1 + B2 * iz2;
    Cq = C0 * iz0 + C1 * iz1 + C2 * iz2;
  }
  size_t cb = (size_t)(b * NCHUNK + chunk) * 256;   // 4 quantities * 64 floats
  const float qa[4] = {A0, A1, A2, Aq};
  const float qb[4] = {B0, B1, B2, Bq};
  const float qc[4] = {C0, C1, C2, Cq};
#pragma unroll
  for (int q = 0; q < 4; ++q) {
    coef[cb + q * 64 + 2 * m + 0]        = qa[q];
    coef[cb + q * 64 + 2 * m + 1]        = qb[q];
    coef[cb + q * 64 + 2 * (16 + m) + 0] = qc[q];
    coef[cb + q * 64 + 2 * (16 + m) + 1] = 0.0f;
  }
  areab[(size_t)(b * NCHUNK + chunk) * 16 + m] = area;
}

// ============ kernel 3: WMMA rasterizer ======================================
__device__ __forceinline__ void stage_copy(const float* gC, const float* gA,
                                           float* lC, float* lA, int tid) {
#pragma unroll
  for (int i = 0; i < 4; ++i) {
    int idx = (tid + i * 256) * 4;           // float index, 16B granules
    cp16(gC + idx, lC + idx);
  }
  if (tid < 64) cp16(gA + tid * 4, lA + tid * 4);
}

__global__ __launch_bounds__(256) void k_raster(const float* __restrict__ coef,
                                                const float* __restrict__ areab,
                                                int* __restrict__ fim) {
  __shared__ __align__(16) float lCoef[2][CH_PER_STAGE * 256];  // 2 x 16 KB
  __shared__ __align__(16) float lArea[2][CH_PER_STAGE * 16];   // 2 x 1 KB

  int b = blockIdx.x / IMG, y = blockIdx.x % IMG;
  int tid = threadIdx.x;
  int wave = tid >> 5, lane = tid & 31;
  int n = lane & 15, hi = lane >> 4;
  int x = wave * 16 + n;
  float px = (2.0f * x + 1.0f) * (1.0f / IMG) - 1.0f;
  float py = 1.0f - (2.0f * y + 1.0f) * (1.0f / IMG);
  // B-matrix (4x16): row0 = 1, row1 = px, row2 = py, row3 = 0
  v2f bm; bm.x = hi ? py : 1.0f; bm.y = hi ? 0.0f : px;

  const float* gC = coef  + (size_t)b * NCHUNK * 256;
  const float* gA = areab + (size_t)b * NCHUNK * 16;

  stage_copy(gC, gA, lCoef[0], lArea[0], tid);
  WAIT_ASYNC();
  __syncthreads();

  float best = FBIG;
  int bidx = -1;

  for (int s = 0; s < NSTAGE; ++s) {
    int d = s & 1;
    if (s + 1 < NSTAGE)
      stage_copy(gC + (size_t)(s + 1) * CH_PER_STAGE * 256,
                 gA + (size_t)(s + 1) * CH_PER_STAGE * 16,
                 lCoef[d ^ 1], lArea[d ^ 1], tid);

    for (int cc = 0; cc < CH_PER_STAGE; ++cc) {
      const float* C = &lCoef[d][cc * 256];
      v2f a0 = *(const v2f*)(C + 0 * 64 + lane * 2);
      v2f a1 = *(const v2f*)(C + 1 * 64 + lane * 2);
      v2f a2 = *(const v2f*)(C + 2 * 64 + lane * 2);
      v2f aq = *(const v2f*)(C + 3 * 64 + lane * 2);
      v8f w0 = wmma4(a0, bm);
      v8f w1 = wmma4(a1, bm);
      v8f w2 = wmma4(a2, bm);
      v8f qq = wmma4(aq, bm);

      const float* AR = &lArea[d][cc * 16 + hi * 8];
      float4 arlo = *(const float4*)(AR);
      float4 arhi = *(const float4*)(AR + 4);
      float areas[8] = {arlo.x, arlo.y, arlo.z, arlo.w,
                        arhi.x, arhi.y, arhi.z, arhi.w};
      int fbase = (s * CH_PER_STAGE + cc) * 16 + hi * 8;
#pragma unroll
      for (int r = 0; r < 8; ++r) {
        float area = areas[r];
        // fast reciprocals (v_rcp_f32); exact IEEE div not needed for a
        // depth-compare.  rcp(0)=inf -> tests fail; invz==0 -> zp=1 (ref rule).
        float invz = qq[r] * __builtin_amdgcn_rcpf(area);
        float zp = (invz == 0.0f) ? 1.0f : __builtin_amdgcn_rcpf(invz);
        bool ok = (w0[r] >= 0.0f) & (w1[r] >= 0.0f) & (w2[r] >= 0.0f) &
                  (area > 0.0f) & (zp > PNEAR) & (zp < PFAR);
        float dpt = ok ? zp : FBIG;                 // cndmask, branch-free
        bool better = dpt < best;
        best = better ? dpt : best;
        bidx = better ? (fbase + r) : bidx;
      }
    }
    if (s + 1 < NSTAGE) {
      WAIT_ASYNC();
      __syncthreads();
    }
  }

  // combine the two C-matrix lane halves (same pixel, face rows 0-7 vs 8-15)
  float ob = __shfl_xor(best, 16, 32);
  int oi = __shfl_xor(bidx, 16, 32);
  if (ob < best || (ob == best && (unsigned)oi < (unsigned)bidx)) {
    best = ob;
    bidx = oi;
  }
  if (hi == 0) fim[((size_t)b * IMG + y) * IMG + x] = bidx;
}

// ============ kernel 4: shading ==============================================
__global__ void k_shade(const int* __restrict__ fim, const float* __restrict__ fvbuf,
                        const float* __restrict__ tex, float* __restrict__ out) {
  int tid = blockIdx.x * blockDim.x + threadIdx.x;
  if (tid >= BB * IMG * IMG) return;
  int b = tid / (IMG * IMG);
  int rem = tid - b * IMG * IMG;
  int y = rem / IMG, x = rem - y * IMG;
  int f = fim[tid];
  float r = 0.f, g = 0.f, bl = 0.f;
  if (f >= 0) {
    int fid = min(f, NF - 1);
    const float* fv = fvbuf + (size_t)(b * NF + fid) * 9;
    float x0 = fv[0], y0 = fv[1], x1 = fv[3], y1 = fv[4], x2 = fv[6], y2 = fv[7];
    float px = (2.0f * x + 1.0f) * (1.0f / IMG) - 1.0f;
    float py = 1.0f - (2.0f * y + 1.0f) * (1.0f / IMG);
    float w0 = (x2 - x1) * (py - y1) - (y2 - y1) * (px - x1);
    float w1 = (x0 - x2) * (py - y2) - (y0 - y2) * (px - x2);
    float w2 = (x1 - x0) * (py - y0) - (y1 - y0) * (px - x0);
    float area = w0 + w1 + w2;
    float sden = (area == 0.0f) ? 1.0f : area;
    float b0 = w0 / sden, b1 = w1 / sden;
    int t0 = min(max((int)floorf(b0 * 3.0f), 0), 2);
    int t1 = min(max((int)floorf(b1 * 3.0f), 0), 2);
    const float* t = tex + ((size_t)(b * NF + fid) * 9 + (t0 * 3 + t1)) * 3;
    r = t[0]; g = t[1]; bl = t[2];
  }
  out[((size_t)(b * 3 + 0) * IMG * IMG) + rem] = r;
  out[((size_t)(b * 3 + 1) * IMG * IMG) + rem] = g;
  out[((size_t)(b * 3 + 2) * IMG * IMG) + rem] = bl;
}

// ============ launcher =======================================================
extern "C" void kernel_launch(void* const* d_in, const int* in_sizes, int n_in,
                              void* d_out, int out_size, void* d_ws, size_t ws_size,
                              hipStream_t stream) {
  const float* cam   = (const float*)d_in[0];
  const float* verts = (const float*)d_in[1];
  const float* uv    = (const float*)d_in[2];
  const float* samp  = (const float*)d_in[3];
  const int*   faces = (const int*)d_in[4];

  float* ws    = (float*)d_ws;
  float* tex   = ws + OFF_TEX;
  float* fvbuf = ws + OFF_FV;
  float* coef  = ws + OFF_COEF;
  float* areab = ws + OFF_AREA;
  int*   fim   = (int*)(ws + OFF_FIM);
  float* out   = (float*)d_out;

  k_tex<<<(BB * NF * 9 + 255) / 256, 256, 0, stream>>>(uv, samp, tex);
  k_face<<<(BB * FPAD + 255) / 256, 256, 0, stream>>>(cam, verts, faces, coef, areab, fvbuf);
  k_raster<<<BB * IMG, 256, 0, stream>>>(coef, areab, fim);
  k_shade<<<(BB * IMG * IMG + 255) / 256, 256, 0, stream>>>(fim, fvbuf, tex, out);
}